// Digit_caps_layer_7009386627584
// MI455X (gfx1250) — compile-verified
//
#include <hip/hip_runtime.h>

// ---------------------------------------------------------------------------
// DigitCaps layer, MI455X (gfx1250).
// Collapses the (buggy, constant-weight) routing loop to one GEMM:
//   s[b, j*16+k] = 0.1 * sum_{i,d} W[i,j,k,d] * x[b,i,d]   (M=256,N=160,K=9216)
//   out = squash(s) over k (16 elems)
// GEMM uses V_WMMA_F32_16X16X4_F32 (full f32 precision, matching reference).
// K is split 4x across gridZ (f32 atomics) and 8x across waves (LDS reduce);
// each wave walks 36 aligned i-groups (8 K each) with pure pointer increments.
// ---------------------------------------------------------------------------

typedef float v2f __attribute__((ext_vector_type(2)));
typedef float v8f __attribute__((ext_vector_type(8)));

#define BATCH   256
#define KTOT    9216          // 1152 * 8
#define NCOL    160           // 10 caps * 16 dims
#define KSPLIT  4             // K split across blockIdx.z (atomic accumulate)
#define WAVES   8             // waves per block (256 threads, wave32)
#define KPW     (KTOT / (KSPLIT * WAVES))   // 288 K per wave
#define IGROUPS (KPW / 8)                   // 36 i-groups (2 WMMAs each)

// ---- zero the f32 accumulator in workspace ---------------------------------
__global__ __launch_bounds__(256) void caps_zero(float* __restrict__ s) {
    int t = blockIdx.x * 256 + threadIdx.x;
    if (t < BATCH * NCOL) s[t] = 0.0f;
}

// ---- main GEMM: one 16x16 C tile per block, K split over waves + gridZ -----
__global__ __launch_bounds__(256) void caps_gemm(const float* __restrict__ x,
                                                 const float* __restrict__ W,
                                                 float* __restrict__ sacc) {
    __shared__ float red[WAVES * 256];

    const int lane = threadIdx.x & 31;
    const int wave = threadIdx.x >> 5;
    const int hi2  = (lane >> 4) << 1;   // 0 for lanes 0-15, 2 for lanes 16-31
    const int lo   = lane & 15;

    const int mbase  = blockIdx.x * 16;
    const int k0wave = (blockIdx.z * WAVES + wave) * KPW;  // multiple of 8

    // A fragment: x[m][K], m = mbase+lo.  Lane's pair = {K+2*hi, K+2*hi+1}
    // (VGPR0 = K0|K2, VGPR1 = K1|K3 per ISA layout) -> contiguous b64 load.
    const float* __restrict__ aptr =
        x + (size_t)(mbase + lo) * KTOT + k0wave + hi2;

    // B fragment: W[i,j,kdim,d], strides (1280,128,8,1) floats.
    //   N = j*16 + kdim (j = blockIdx.y, kdim = lo), K = i*8 + d.
    // Lane's pair differs only in d -> d-adjacent in memory -> b64 load.
    const float* __restrict__ bptr =
        W + (size_t)blockIdx.y * 128 + (size_t)lo * 8
          + (size_t)(k0wave >> 3) * 1280 + hi2;

    v8f c = {};  // 16x16 f32 accumulator (8 VGPRs)

    #pragma unroll 2
    for (int g = 0; g < IGROUPS; ++g) {
        // Two K-chunks of 4 inside one i-group: byte offsets 0 and 16.
        v2f a0 = *(const v2f*)(aptr);
        v2f b0 = *(const v2f*)(bptr);
        v2f a1 = *(const v2f*)(aptr + 4);
        v2f b1 = *(const v2f*)(bptr + 4);
        // (neg_a, A, neg_b, B, c_mod, C, reuse_a, reuse_b)
        c = __builtin_amdgcn_wmma_f32_16x16x4_f32(
                false, a0, false, b0, (short)0, c, false, false);
        c = __builtin_amdgcn_wmma_f32_16x16x4_f32(
                false, a1, false, b1, (short)0, c, false, false);
        aptr += 8;      //  32 B: next i-group along x's K axis
        bptr += 1280;   // 5120 B: next prev-capsule row block of W
    }

    // Cross-wave reduction of the 8 partial C tiles through LDS.
    #pragma unroll
    for (int r = 0; r < 8; ++r)
        red[wave * 256 + r * 32 + lane] = c[r];
    __syncthreads();

    const int t = threadIdx.x;               // t = r*32 + l in wave layout
    float sum = 0.0f;
    #pragma unroll
    for (int w = 0; w < WAVES; ++w)
        sum += red[w * 256 + t];

    // C/D layout: VGPR r, lane l -> row = r + 8*(l>>4), col = l & 15
    const int r   = t >> 5;
    const int l   = t & 31;
    const int row = r + 8 * (l >> 4);
    const int col = l & 15;
    const int m   = mbase + row;
    const int n   = blockIdx.y * 16 + col;

    atomicAdd(&sacc[(size_t)m * NCOL + n], sum);   // K-split accumulation
}

// ---- squash: scale by routing weight 0.1, normalize over 16-dim capsule ----
__global__ __launch_bounds__(256) void caps_squash(const float* __restrict__ sacc,
                                                   float* __restrict__ out) {
    int t = blockIdx.x * 256 + threadIdx.x;        // flat [b, j, k], k fastest
    float v = 0.1f * sacc[t];                      // c = softmax(0) = 1/10

    // sum of squares over the 16-lane capsule group (masks <=8 stay in-group)
    float sq = v * v;
    sq += __shfl_xor(sq, 1, 32);
    sq += __shfl_xor(sq, 2, 32);
    sq += __shfl_xor(sq, 4, 32);
    sq += __shfl_xor(sq, 8, 32);

    float nrm = sqrtf(sq + 1e-7f);
    out[t] = (sq / (1.0f + sq)) * (v / nrm);
}

// ---------------------------------------------------------------------------
extern "C" void kernel_launch(void* const* d_in, const int* in_sizes, int n_in,
                              void* d_out, int out_size, void* d_ws, size_t ws_size,
                              hipStream_t stream) {
    const float* x = (const float*)d_in[0];   // [256, 1152, 8] f32
    const float* W = (const float*)d_in[1];   // [1, 1152, 10, 16, 8] f32
    float* out  = (float*)d_out;              // [256, 1, 10, 16, 1] f32 (40960)
    float* sacc = (float*)d_ws;               // 256*160 f32 accumulator

    const int nflat = BATCH * NCOL;           // 40960
    caps_zero  <<<nflat / 256, 256, 0, stream>>>(sacc);
    caps_gemm  <<<dim3(16, 10, KSPLIT), 256, 0, stream>>>(x, W, sacc);
    caps_squash<<<nflat / 256, 256, 0, stream>>>(sacc, out);
}